// Entropy_Hist_65094524338632
// MI455X (gfx1250) — compile-verified
//
#include <hip/hip_runtime.h>
#include <hip/hip_bf16.h>
#include <stdint.h>

typedef __attribute__((ext_vector_type(2))) float v2f;
typedef __attribute__((ext_vector_type(8))) float v8f;

// ---------------- order-preserving float <-> uint keys (no NaNs in data) ----
__device__ __forceinline__ unsigned flipf(float f) {
  unsigned u = __float_as_uint(f);
  return u ^ ((u >> 31) ? 0xFFFFFFFFu : 0x80000000u);
}
__device__ __forceinline__ float unflipf(unsigned u) {
  unsigned v = u ^ ((u >> 31) ? 0x80000000u : 0xFFFFFFFFu);
  return __uint_as_float(v);
}

// ---------------- CDNA5 async global<->LDS DMA helpers ----------------------
__device__ __forceinline__ void async_ld_b128(unsigned lds_off, unsigned long long gaddr) {
  asm volatile("global_load_async_to_lds_b128 %0, %1, off"
               :: "v"(lds_off), "v"(gaddr) : "memory");
}
__device__ __forceinline__ void async_st_b128(unsigned long long gaddr, unsigned lds_off) {
  asm volatile("global_store_async_from_lds_b128 %0, %1, off"
               :: "v"(gaddr), "v"(lds_off) : "memory");
}
__device__ __forceinline__ void wait_async0() {
  asm volatile("s_wait_asynccnt 0" ::: "memory");
}

// ---------------- init ------------------------------------------------------
__global__ __launch_bounds__(256) void init_kernel(unsigned* minmax, unsigned* hist) {
  int i = blockIdx.x * 256 + threadIdx.x;   // grid 128 -> exactly 32768 bins
  hist[i] = 0u;
  if (i == 0) { minmax[0] = 0xFFFFFFFFu; minmax[1] = 0u; }
}

// ---------------- stencil tile (exact reference op order) -------------------
// outputs z = z0 .. z0+NOUT-1, reads row elements z0-1 .. z0+NOUT (EXTRA: +2 scalars)
template <int NOUT, bool EXTRA, int PASS>
__device__ __forceinline__ void do_tile(const float* __restrict__ base, int h, int w, int z0,
                                        float& lmin, float& lmax,
                                        unsigned* myhist, float mn, float range) {
  float s[NOUT], ctr[NOUT];
#pragma unroll
  for (int i = 0; i < NOUT; ++i) s[i] = 0.0f;
#pragma unroll
  for (int di = 0; di < 3; ++di) {
#pragma unroll
    for (int dj = 0; dj < 3; ++dj) {
      const float* r = base + (((h - 1 + di) << 6) + (w - 1 + dj)) * 64 + (z0 - 1);
      float rv[18];
      const float4* r4 = (const float4*)r;                 // 16B aligned (z0-1 % 16 == 0)
      float4 a = r4[0], b = r4[1], c = r4[2], d = r4[3];
      rv[0]=a.x; rv[1]=a.y; rv[2]=a.z; rv[3]=a.w;
      rv[4]=b.x; rv[5]=b.y; rv[6]=b.z; rv[7]=b.w;
      rv[8]=c.x; rv[9]=c.y; rv[10]=c.z; rv[11]=c.w;
      rv[12]=d.x; rv[13]=d.y; rv[14]=d.z; rv[15]=d.w;
      if (EXTRA) { rv[16] = r[16]; rv[17] = r[17]; } else { rv[16] = 0.f; rv[17] = 0.f; }
#pragma unroll
      for (int z = 0; z < NOUT; ++z) {
        // exact reference order: the three dk terms added consecutively per (di,dj)
        s[z] = __fadd_rn(s[z], rv[z]);
        s[z] = __fadd_rn(s[z], rv[z + 1]);
        s[z] = __fadd_rn(s[z], rv[z + 2]);
      }
      if (di == 1 && dj == 1) {
#pragma unroll
        for (int z = 0; z < NOUT; ++z) ctr[z] = rv[z + 1];
      }
    }
  }
#pragma unroll
  for (int z = 0; z < NOUT; ++z) {
    float c = ctr[z];
    float mean_p = __fdiv_rn(__fsub_rn(s[z], c), 26.0f);
    float ij = __fadd_rn(__fmul_rn(c, 100.0f), mean_p);    // no fma contraction
    if (PASS == 0) {
      lmin = fminf(lmin, ij);
      lmax = fmaxf(lmax, ij);
    } else {
      float t = __fdiv_rn(__fsub_rn(ij, mn), range);
      float f = floorf(__fmul_rn(t, 256.0f));
      f = fminf(fmaxf(f, 0.0f), 255.0f);
      atomicAdd(&myhist[(int)f], 1u);
    }
  }
}

// ---------------- stencil passes (PASS 0 = min/max, PASS 1 = histogram) -----
// 8 lane-interleaved sub-histograms, 257-dword stride (bank decorrelation).
template <int PASS>
__global__ __launch_bounds__(256) void stencil_kernel(const float* __restrict__ img,
                                                      unsigned* __restrict__ minmax,
                                                      unsigned* __restrict__ hist) {
  __shared__ unsigned shist[8 * 257];
  __shared__ unsigned smin, smax;
  const int ch = blockIdx.x;                 // 0..127
  const float* base = img + ((size_t)ch << 18);

  if (PASS == 1) {
    for (int i = threadIdx.x; i < 8 * 257; i += 256) shist[i] = 0u;
  } else if (threadIdx.x == 0) {
    smin = 0xFFFFFFFFu; smax = 0u;
  }
  __syncthreads();

  unsigned* myhist = shist + (threadIdx.x & 7) * 257;

  float mn = 0.f, range = 1.f;
  if (PASS == 1) {
    float lo = unflipf(minmax[0]);
    float hi = unflipf(minmax[1]);
    mn = lo;
    range = __fsub_rn(hi, lo);
  }

  float lmin = INFINITY, lmax = -INFINITY;
  for (int r = blockIdx.y * blockDim.x + threadIdx.x; r < 62 * 62;
       r += gridDim.y * blockDim.x) {
    int h = 1 + r / 62;
    int w = 1 + r % 62;
    do_tile<16, true,  PASS>(base, h, w, 1,  lmin, lmax, myhist, mn, range);
    do_tile<16, true,  PASS>(base, h, w, 17, lmin, lmax, myhist, mn, range);
    do_tile<16, true,  PASS>(base, h, w, 33, lmin, lmax, myhist, mn, range);
    do_tile<14, false, PASS>(base, h, w, 49, lmin, lmax, myhist, mn, range);
  }

  if (PASS == 0) {
    atomicMin(&smin, flipf(lmin));
    atomicMax(&smax, flipf(lmax));
    __syncthreads();
    if (threadIdx.x == 0) {
      atomicMin(&minmax[0], smin);
      atomicMax(&minmax[1], smax);
    }
  } else {
    __syncthreads();
    unsigned total = 0u;
#pragma unroll
    for (int k = 0; k < 8; ++k) total += shist[k * 257 + threadIdx.x];
    atomicAdd(&hist[(ch << 8) + threadIdx.x], total);
  }
}

// ---------------- entropy per channel: WMMA-based 256-term reduction --------
// D = A(16x4, all ones) x B(4x16, packed h-values) + C  =>  each row of D holds
// the column sums of B; every B element contributes exactly once, so the total
// (sum of one D row across 16 lanes) is independent of the B packing bijection.
__global__ __launch_bounds__(32) void entropy_kernel(const unsigned* __restrict__ hist,
                                                     float* __restrict__ ent) {
  int ch = blockIdx.x;
  int l = threadIdx.x;                       // one wave32 per channel
  float h[8];
#pragma unroll
  for (int i = 0; i < 8; ++i) {
    float cnt = (float)hist[(ch << 8) + (i << 5) + l];
    float p = __fdiv_rn(cnt, 287496.0f);     // (H+2)(W+2)(Z+2) = 66^3
    float lg = logf(fmaxf(p, 1e-40f));
    h[i] = __fdiv_rn(__fmul_rn(-p, lg), 0.69314718055994530942f);
  }
  float v;
#if __has_builtin(__builtin_amdgcn_wmma_f32_16x16x4_f32)
  v8f c = {};
  v2f a; a.x = 1.0f; a.y = 1.0f;             // all-ones A matrix (16x4)
#pragma unroll
  for (int i = 0; i < 4; ++i) {
    v2f b; b.x = h[2 * i]; b.y = h[2 * i + 1];  // 64 h-values per WMMA
    c = __builtin_amdgcn_wmma_f32_16x16x4_f32(false, a, false, b,
                                              (short)0, c, false, false);
  }
  v = c[0];                                  // row 0 column-sums live in lanes 0..15
  v += __shfl_down(v, 8, 16);
  v += __shfl_down(v, 4, 16);
  v += __shfl_down(v, 2, 16);
  v += __shfl_down(v, 1, 16);
#else
  v = ((h[0] + h[1]) + (h[2] + h[3])) + ((h[4] + h[5]) + (h[6] + h[7]));
  v += __shfl_down(v, 16, 32);
  v += __shfl_down(v, 8, 32);
  v += __shfl_down(v, 4, 32);
  v += __shfl_down(v, 2, 32);
  v += __shfl_down(v, 1, 32);
#endif
  if (l == 0) ent[ch] = v;
}

// ---------------- top-k (descending value, ties -> lowest index) ------------
__global__ void topk_kernel(const float* __restrict__ ent, int* __restrict__ idxo) {
  int b = blockIdx.x;
  if (threadIdx.x != 0) return;
  unsigned long long used = 0ull;
  for (int j = 0; j < 32; ++j) {
    float bv = -INFINITY; int bi = 0;
    for (int c = 0; c < 64; ++c) {
      if (used & (1ull << c)) continue;
      float e = ent[b * 64 + c];
      if (e > bv) { bv = e; bi = c; }                  // strict > == jax tie-break
    }
    used |= 1ull << bi;
    idxo[b * 32 + j] = bi;
  }
}

// ---------------- gather selected channels via async LDS DMA ----------------
// 2048 blocks: each copies a 32KB chunk. Wave-private LDS regions, so only
// per-wave s_wait_asynccnt between the load and store phases is needed.
__global__ __launch_bounds__(256) void gather_kernel(const float* __restrict__ img,
                                                     const int* __restrict__ idx,
                                                     float* __restrict__ out) {
  __shared__ __align__(16) char smem[32768];
  int blk = blockIdx.x;
  int chan = blk >> 5;            // 0..63  (b*32 + j)
  int part = blk & 31;            // 32KB chunk within the 1MiB channel
  int b = chan >> 5, j = chan & 31;
  int sc = idx[(b << 5) + j];

  unsigned long long src = (unsigned long long)(uintptr_t)img
                         + (((unsigned long long)((b << 6) + sc)) << 20)   // channel * 1MiB
                         + ((unsigned long long)part << 15)
                         + (unsigned long long)threadIdx.x * 128ull;
  unsigned long long dst = (unsigned long long)(uintptr_t)out
                         + ((unsigned long long)chan << 20)
                         + ((unsigned long long)part << 15)
                         + (unsigned long long)threadIdx.x * 128ull;
  unsigned lbase = (unsigned)(uintptr_t)smem + threadIdx.x * 128u;  // low 32b = LDS offset

#pragma unroll
  for (int i = 0; i < 8; ++i) async_ld_b128(lbase + i * 16u, src + (unsigned long long)(i * 16));
  wait_async0();
#pragma unroll
  for (int i = 0; i < 8; ++i) async_st_b128(dst + (unsigned long long)(i * 16), lbase + i * 16u);
  wait_async0();
}

// ---------------- launch ----------------------------------------------------
extern "C" void kernel_launch(void* const* d_in, const int* in_sizes, int n_in,
                              void* d_out, int out_size, void* d_ws, size_t ws_size,
                              hipStream_t stream) {
  const float* img = (const float*)d_in[0];      // (2,64,64,64,64) f32
  float* out = (float*)d_out;                    // (2,32,64,64,64) f32

  char* ws = (char*)d_ws;
  unsigned* minmax = (unsigned*)ws;                          // 2 u32
  unsigned* hist   = (unsigned*)(ws + 256);                  // 128*256 u32 = 128KB
  float*    ent    = (float*)   (ws + 256 + 131072);         // 128 f32
  int*      idxo   = (int*)     (ws + 256 + 131072 + 512);   // 64 i32

  init_kernel<<<128, 256, 0, stream>>>(minmax, hist);
  stencil_kernel<0><<<dim3(128, 4), 256, 0, stream>>>(img, minmax, hist);
  stencil_kernel<1><<<dim3(128, 4), 256, 0, stream>>>(img, minmax, hist);
  entropy_kernel<<<128, 32, 0, stream>>>(hist, ent);
  topk_kernel<<<2, 64, 0, stream>>>(ent, idxo);
  gather_kernel<<<2048, 256, 0, stream>>>(img, idxo, out);
}